// ECModel_85856396248062
// MI455X (gfx1250) — compile-verified
//
#include <hip/hip_runtime.h>
#include <hip/hip_bf16.h>

// Native clang vectors (needed for __builtin_nontemporal_load on 128-bit data).
typedef int   v4i __attribute__((ext_vector_type(4)));
typedef float v4f __attribute__((ext_vector_type(4)));

#define EC_AS1 __attribute__((address_space(1)))
#define EC_AS3 __attribute__((address_space(3)))

#define EC_BLOCK 256              // 8 wave32s per block
#define EC_WAVES_PER_BLOCK 8
#define EC_TILE 128               // edges per wave per tile (32 lanes x 4)
#define EC_MAX_BLOCKS 4096        // cap -> 32768 waves, ~7.6 tiles/wave at E=32M

// ---------------------------------------------------------------------------
// gfx1250 async LDS-staging helpers.
// One 16B async load per lane: global -> LDS, tracked by ASYNCcnt (no VGPR
// writeback, frees the load-return path for the random gathers/atomics).
// Falls back to a synchronous NT copy if the toolchain lacks the builtin.
// ---------------------------------------------------------------------------
__device__ __forceinline__ void ec_async_ld16(const int* g, int* l) {
#if __has_builtin(__builtin_amdgcn_global_load_async_to_lds_b128)
  __builtin_amdgcn_global_load_async_to_lds_b128(
      (EC_AS1 v4i*)(unsigned long long)g,
      (EC_AS3 v4i*)(unsigned)(unsigned long long)l,
      0, 0);
#else
  *(v4i*)l = __builtin_nontemporal_load((const v4i*)g);
#endif
}

template <int N>
__device__ __forceinline__ void ec_wait_async() {
#if __has_builtin(__builtin_amdgcn_s_wait_asynccnt)
  __builtin_amdgcn_s_wait_asynccnt(N);
#elif __has_builtin(__builtin_amdgcn_global_load_async_to_lds_b128)
  asm volatile("s_wait_asynccnt %0" ::"i"(N) : "memory");
#endif
  asm volatile("" ::: "memory");   // keep LDS reads below the wait
}

__device__ __forceinline__ v4i ec_lds_read16(const int* p) {
  return *(const EC_AS3 v4i*)(unsigned)(unsigned long long)p;
}

// ---------------------------------------------------------------------------
// Zero both accumulators (agg1 in workspace, agg2 == d_out) each call.
// ---------------------------------------------------------------------------
__global__ void ec_zero2(float* __restrict__ a, float* __restrict__ b, int n) {
  int i4 = (blockIdx.x * blockDim.x + threadIdx.x) << 2;
  if (i4 + 3 < n) {
    v4f z = {0.f, 0.f, 0.f, 0.f};
    *(v4f*)(a + i4) = z;
    *(v4f*)(b + i4) = z;
  } else {
    for (int k = i4; k < n; ++k) { a[k] = 0.f; b[k] = 0.f; }
  }
}

// ---------------------------------------------------------------------------
// Edge scatter pass with async double-buffered index staging:
//   agg[dst[e]] += f(x[src[e]]),  f = relu for pass 2.
// Per wave: a private 2 KB LDS slice holding two (128 src + 128 dst) tiles.
// Pipeline: issue tile k+1 (2 async b128/wave) -> s_wait_asynccnt 2 ->
// ds_load_b128 tile k -> 4 gathers + 4 global_atomic_add_f32 per lane.
// Gathers/atomics hit the L2-resident 4 MB node arrays (192 MB L2).
// ---------------------------------------------------------------------------
template <bool RELU>
__global__ __launch_bounds__(EC_BLOCK)
void ec_scatter_async(const float* __restrict__ x,
                      const int*   __restrict__ src,
                      const int*   __restrict__ dst,
                      float*       __restrict__ agg,
                      int nTiles, int tileStride) {
  __shared__ int smem[EC_WAVES_PER_BLOCK * 2 * 2 * EC_TILE];  // 16 KB / block
  const int lane = threadIdx.x & 31;
  const int wIn  = threadIdx.x >> 5;

  int t = blockIdx.x * EC_WAVES_PER_BLOCK + wIn;   // wave-uniform
  if (t >= nTiles) return;

  int* wbase = smem + wIn * (2 * 2 * EC_TILE);     // this wave's 512-int slice
  const int lo = lane << 2;                        // lane's 4-int offset

  // Stage first tile into buffer 0.
  {
    const size_t e = ((size_t)t << 7) + lo;
    ec_async_ld16(src + e, wbase + lo);
    ec_async_ld16(dst + e, wbase + EC_TILE + lo);
  }

  int buf = 0;
  for (;;) {
    const int tn = t + tileStride;
    if (tn < nTiles) {
      const int o = ((buf ^ 1) << 8) + lo;
      const size_t e = ((size_t)tn << 7) + lo;
      ec_async_ld16(src + e, wbase + o);
      ec_async_ld16(dst + e, wbase + o + EC_TILE);
      ec_wait_async<2>();          // tile t's 2 loads complete (in-order)
    } else {
      ec_wait_async<0>();
    }

    const int o = (buf << 8) + lo;
    v4i s = ec_lds_read16(wbase + o);             // ds_load_b128
    v4i d = ec_lds_read16(wbase + o + EC_TILE);   // ds_load_b128

    float v0 = x[s.x];
    float v1 = x[s.y];
    float v2 = x[s.z];
    float v3 = x[s.w];
    if (RELU) {
      v0 = fmaxf(v0, 0.f); v1 = fmaxf(v1, 0.f);
      v2 = fmaxf(v2, 0.f); v3 = fmaxf(v3, 0.f);
    }
    atomicAdd(agg + d.x, v0);      // global_atomic_add_f32 (no return)
    atomicAdd(agg + d.y, v1);
    atomicAdd(agg + d.z, v2);
    atomicAdd(agg + d.w, v3);

    if (tn >= nTiles) break;
    t = tn;
    buf ^= 1;
  }
}

// Scalar path: tile remainder and the (never-expected) unaligned case.
template <bool RELU>
__global__ void ec_scatter_scalar(const float* __restrict__ x,
                                  const int*   __restrict__ src,
                                  const int*   __restrict__ dst,
                                  float*       __restrict__ agg,
                                  int nE) {
  int e = blockIdx.x * blockDim.x + threadIdx.x;
  if (e < nE) {
    float v = x[src[e]];
    if (RELU) v = fmaxf(v, 0.f);
    atomicAdd(agg + dst[e], v);
  }
}

// ---------------------------------------------------------------------------
// Final node pass, in place on d_out: out = sigmoid(relu(agg2)).
// ---------------------------------------------------------------------------
__global__ void ec_relu_sigmoid(float* __restrict__ io, int n) {
  int i4 = (blockIdx.x * blockDim.x + threadIdx.x) << 2;
  if (i4 + 3 < n) {
    v4f v = *(const v4f*)(io + i4);
    v4f r;
#pragma unroll
    for (int k = 0; k < 4; ++k) {
      float h = fmaxf(v[k], 0.f);
      r[k] = 1.f / (1.f + __expf(-h));
    }
    *(v4f*)(io + i4) = r;
  } else {
    for (int k = i4; k < n; ++k) {
      float h = fmaxf(io[k], 0.f);
      io[k] = 1.f / (1.f + __expf(-h));
    }
  }
}

// ---------------------------------------------------------------------------
// Launcher.
// Inputs (setup_inputs order): d_in[0] = x [N,1] f32
//                              d_in[1] = edge_index [2,E] int (row0=src,row1=dst)
//                              d_in[2] = w1 (1x1 identity, unused)
//                              d_in[3] = w2 (1x1 identity, unused)
// d_out: N f32 (doubles as second accumulator).  d_ws: >= N*4 bytes (agg1).
// ---------------------------------------------------------------------------
extern "C" void kernel_launch(void* const* d_in, const int* in_sizes, int n_in,
                              void* d_out, int out_size, void* d_ws, size_t ws_size,
                              hipStream_t stream) {
  const float* x    = (const float*)d_in[0];
  const int*   edge = (const int*)d_in[1];

  const int N = in_sizes[0];        // x is [N,1] -> N elements
  const int E = in_sizes[1] / 2;    // edge_index is [2,E]

  const int* src = edge;            // row 0
  const int* dst = edge + E;        // row 1

  float* agg1 = (float*)d_ws;       // h1 accumulator (relu applied at gather)
  float* out  = (float*)d_out;      // h2 accumulator, then sigmoid'd in place

  const dim3 blk(EC_BLOCK);
  const int gridN4 = ((N + 3) / 4 + EC_BLOCK - 1) / EC_BLOCK;

  // 1) zero both accumulators (harness poisons buffers; must be deterministic)
  ec_zero2<<<gridN4, blk, 0, stream>>>(agg1, out, N);

  const int F = E >> 7;             // full 128-edge tiles
  if (((E & 3) == 0) && F > 0) {    // 16B alignment of dst stream requires E%4==0
    int blocks = (F + 2 * EC_WAVES_PER_BLOCK - 1) / (2 * EC_WAVES_PER_BLOCK);
    if (blocks > EC_MAX_BLOCKS) blocks = EC_MAX_BLOCKS;
    const int stride = blocks * EC_WAVES_PER_BLOCK;
    const int rem    = E - (F << 7);
    const int e0     = F << 7;

    // pass 1: agg1[dst] += x[src]
    ec_scatter_async<false><<<blocks, blk, 0, stream>>>(x, src, dst, agg1, F, stride);
    if (rem)
      ec_scatter_scalar<false><<<(rem + EC_BLOCK - 1) / EC_BLOCK, blk, 0, stream>>>(
          x, src + e0, dst + e0, agg1, rem);

    // pass 2: out[dst] += relu(agg1[src])
    ec_scatter_async<true><<<blocks, blk, 0, stream>>>(agg1, src, dst, out, F, stride);
    if (rem)
      ec_scatter_scalar<true><<<(rem + EC_BLOCK - 1) / EC_BLOCK, blk, 0, stream>>>(
          agg1, src + e0, dst + e0, out, rem);
  } else {
    const int gridE = (E + EC_BLOCK - 1) / EC_BLOCK;
    ec_scatter_scalar<false><<<gridE, blk, 0, stream>>>(x,    src, dst, agg1, E);
    ec_scatter_scalar<true ><<<gridE, blk, 0, stream>>>(agg1, src, dst, out,  E);
  }

  // 4) out = sigmoid(relu(out))
  ec_relu_sigmoid<<<gridN4, blk, 0, stream>>>(out, N);
}